// GraphConvEdgeLengthWeighted_25159918420161
// MI455X (gfx1250) — compile-verified
//
#include <hip/hip_runtime.h>
#include <hip/hip_bf16.h>

typedef __attribute__((ext_vector_type(2))) float v2f;
typedef __attribute__((ext_vector_type(8))) float v8f;

#define GC_N      8192
#define GC_F      16384
#define GC_D      64
#define GC_EPS    1e-7f
#define HASH_BITS 18
#define HASH_SZ   (1u << HASH_BITS)
#define HASH_MASK (HASH_SZ - 1u)
#define EMPTY_KEY 0xFFFFFFFFu

#if __has_builtin(__builtin_amdgcn_global_load_async_to_lds_b128)
#define GC_HAS_ASYNC_LDS 1
typedef int gc_v4i __attribute__((vector_size(16)));
typedef __attribute__((address_space(1))) gc_v4i* gc_gvp;  // global (device) int4*
typedef __attribute__((address_space(3))) gc_v4i* gc_lvp;  // LDS int4*
#else
#define GC_HAS_ASYNC_LDS 0
#endif

// ---------------------------------------------------------------------------
// Workspace layout (bytes):
//   [0,              1048576)  hash table, uint32[HASH_SZ]
//   [1048576,        1081344)  colsum delta, float[N]
//   [1081344,        1081600)  S_total, float[64]
//   [1081600,        3178752)  acc, float[N*64]
// ---------------------------------------------------------------------------

__global__ void gc_init_ws(unsigned* __restrict__ hash,
                           float* __restrict__ colsum,
                           float* __restrict__ S,
                           float* __restrict__ acc) {
    const int stride = gridDim.x * blockDim.x;
    int tid = blockIdx.x * blockDim.x + threadIdx.x;
    for (int i = tid; i < (int)HASH_SZ; i += stride) hash[i] = EMPTY_KEY;
    for (int i = tid; i < GC_N; i += stride) colsum[i] = 0.0f;
    for (int i = tid; i < GC_D; i += stride) S[i] = 0.0f;
    for (int i = tid; i < GC_N * GC_D; i += stride) acc[i] = 0.0f;
}

// S_total[k] = sum over all rows of input[:,k]
__global__ void gc_colsum_S(const float* __restrict__ input, float* __restrict__ S) {
    const int k = threadIdx.x;        // 0..63
    const int chunk = blockIdx.x;     // 64 chunks of 128 rows
    float s = 0.0f;
    const int r0 = chunk * (GC_N / 64);
    #pragma unroll 4
    for (int r = 0; r < GC_N / 64; ++r)
        s += input[(r0 + r) * GC_D + k];
    atomicAdd(&S[k], s);
}

// One thread per directed edge instance (6 per face). Dedupe via hash table;
// winner accumulates colsum[dst] += wp and acc[src][:] += wp * input[dst][:],
// where wp = max(w, EPS) - EPS.
__global__ void gc_edges(const int* __restrict__ faces,
                         const float* __restrict__ verts,
                         const float* __restrict__ sigma,
                         const float* __restrict__ input,
                         unsigned* __restrict__ hash,
                         float* __restrict__ colsum,
                         float* __restrict__ acc) {
    const int tid = blockIdx.x * blockDim.x + threadIdx.x;
    if (tid >= 6 * GC_F) return;
    const int f = tid / 6;
    const int e = tid - 6 * f;
    // edges = [(0,1),(0,2),(1,2),(1,0),(2,0),(2,1)]
    const int ea[6] = {0, 0, 1, 1, 2, 2};
    const int eb[6] = {1, 2, 2, 0, 0, 1};
    const int src = faces[f * 3 + ea[e]];
    const int dst = faces[f * 3 + eb[e]];

    const float dx = verts[src * 3 + 0] - verts[dst * 3 + 0];
    const float dy = verts[src * 3 + 1] - verts[dst * 3 + 1];
    const float dz = verts[src * 3 + 2] - verts[dst * 3 + 2];
    const float d2 = dx * dx + dy * dy + dz * dz;
    const float sg = sigma[0];
    const float w = expf(-d2 / (sg * sg));
    const float wp = fmaxf(w, GC_EPS) - GC_EPS;

    const unsigned key = ((unsigned)src << 13) | (unsigned)dst;
    unsigned h = (key * 2654435761u) >> (32 - HASH_BITS);
    bool winner = false;
    for (unsigned probe = 0; probe < HASH_SZ; ++probe) {
        const unsigned idx = (h + probe) & HASH_MASK;
        const unsigned old = atomicCAS(&hash[idx], EMPTY_KEY, key);
        if (old == EMPTY_KEY) { winner = true; break; }
        if (old == key) break;  // duplicate edge: .set semantics -> count once
    }
    if (!winner) return;

    atomicAdd(&colsum[dst], wp);
    const float* __restrict__ in_d = input + dst * GC_D;
    float* __restrict__ acc_s = acc + src * GC_D;
    #pragma unroll 8
    for (int k = 0; k < GC_D; ++k)
        atomicAdd(&acc_s[k], wp * in_d[k]);
}

// ---------------------------------------------------------------------------
// Fused output GEMM: out = input @ fc_w^T + fc_b
//                        + (dinv .* (EPS*S + acc)) @ nfc_w^T + nfc_b
//
// Block = 128 threads (4 wave32), one 16-row stripe of out (16x64).
// Stage input tile / acc tile / both weight matrices / S into LDS (41 KB of
// the 320 KB WGP pool) via async global->LDS copies (ASYNCcnt), then each
// wave computes one 16x16 fp32 tile with V_WMMA_F32_16X16X4_F32
// (16 K-steps x 2 GEMMs = 32 WMMAs per wave), reading fragments from LDS.
// ---------------------------------------------------------------------------

// LDS float offsets
#define L_IN   0        // 16*64
#define L_ACC  1024     // 16*64
#define L_W1   2048     // 64*64
#define L_W2   6144     // 64*64
#define L_S    10240    // 64
#define L_TOT  10304    // floats (41216 bytes)

__global__ __launch_bounds__(128) void gc_fused_gemm(
    const float* __restrict__ input,
    const float* __restrict__ fc_w, const float* __restrict__ fc_b,
    const float* __restrict__ nfc_w, const float* __restrict__ nfc_b,
    const float* __restrict__ colsum, const float* __restrict__ S,
    const float* __restrict__ acc, float* __restrict__ out) {
    __shared__ float smem[L_TOT];

    const int lane = threadIdx.x & 31;
    const int wv   = threadIdx.x >> 5;       // wave id 0..3 -> col tile
    const int m0   = blockIdx.x * 16;        // row tile
    const int c0   = wv * 16;                // col tile
    const int nn   = lane & 15;              // N (and A's M) position
    const int kh   = lane >> 4;              // K half: K = 2*kh + v

    const int row  = m0 + nn;                // A-fragment row for this lane
    const int colB = c0 + nn;                // B-fragment column for this lane

    // ---- Stage to LDS (16-byte granules, 128 threads cooperatively) ----
    {
        const float* __restrict__ gin  = input + m0 * GC_D;
        const float* __restrict__ gacc = acc + m0 * GC_D;
        const int t = threadIdx.x;
#if GC_HAS_ASYNC_LDS
        #pragma unroll
        for (int i = t; i < 1024 / 4; i += 128) {
            __builtin_amdgcn_global_load_async_to_lds_b128(
                (gc_gvp)(gin + i * 4),  (gc_lvp)(smem + L_IN  + i * 4), 0, 0);
            __builtin_amdgcn_global_load_async_to_lds_b128(
                (gc_gvp)(gacc + i * 4), (gc_lvp)(smem + L_ACC + i * 4), 0, 0);
        }
        #pragma unroll
        for (int i = t; i < 4096 / 4; i += 128) {
            __builtin_amdgcn_global_load_async_to_lds_b128(
                (gc_gvp)(fc_w + i * 4),  (gc_lvp)(smem + L_W1 + i * 4), 0, 0);
            __builtin_amdgcn_global_load_async_to_lds_b128(
                (gc_gvp)(nfc_w + i * 4), (gc_lvp)(smem + L_W2 + i * 4), 0, 0);
        }
        if (t < 16)
            __builtin_amdgcn_global_load_async_to_lds_b128(
                (gc_gvp)(S + t * 4), (gc_lvp)(smem + L_S + t * 4), 0, 0);
        asm volatile("s_wait_asynccnt 0x0" ::: "memory");
#else
        #pragma unroll
        for (int i = t; i < 1024 / 4; i += 128) {
            ((float4*)(smem + L_IN))[i]  = ((const float4*)gin)[i];
            ((float4*)(smem + L_ACC))[i] = ((const float4*)gacc)[i];
        }
        #pragma unroll
        for (int i = t; i < 4096 / 4; i += 128) {
            ((float4*)(smem + L_W1))[i] = ((const float4*)fc_w)[i];
            ((float4*)(smem + L_W2))[i] = ((const float4*)nfc_w)[i];
        }
        if (t < 16)
            ((float4*)(smem + L_S))[t] = ((const float4*)S)[t];
#endif
    }
    __syncthreads();

    // C init = broadcast bias per output column
    const float bias = fc_b[colB] + nfc_b[colB];
    v8f d;
    #pragma unroll
    for (int v = 0; v < 8; ++v) d[v] = bias;

    // GEMM 1: input @ fc_w^T, fragments from LDS
    const float* __restrict__ sA1 = smem + L_IN + nn * GC_D;
    const float* __restrict__ sB1 = smem + L_W1 + colB * GC_D;
    #pragma unroll
    for (int kk = 0; kk < GC_D; kk += 4) {
        const int k0 = kk + 2 * kh;
        v2f a; a.x = sA1[k0]; a.y = sA1[k0 + 1];
        v2f b; b.x = sB1[k0]; b.y = sB1[k0 + 1];
        d = __builtin_amdgcn_wmma_f32_16x16x4_f32(false, a, false, b,
                                                  (short)0, d, false, false);
    }

    // GEMM 2: (dinv * (EPS*S + acc)) @ nfc_w^T  -- A2 built on the fly
    const float dinv = 1.0f / ((float)GC_N * GC_EPS + colsum[row]);
    const float* __restrict__ sA2 = smem + L_ACC + nn * GC_D;
    const float* __restrict__ sB2 = smem + L_W2 + colB * GC_D;
    const float* __restrict__ sS  = smem + L_S;
    #pragma unroll
    for (int kk = 0; kk < GC_D; kk += 4) {
        const int k0 = kk + 2 * kh;
        v2f a;
        a.x = dinv * fmaf(GC_EPS, sS[k0],     sA2[k0]);
        a.y = dinv * fmaf(GC_EPS, sS[k0 + 1], sA2[k0 + 1]);
        v2f b; b.x = sB2[k0]; b.y = sB2[k0 + 1];
        d = __builtin_amdgcn_wmma_f32_16x16x4_f32(false, a, false, b,
                                                  (short)0, d, false, false);
    }

    // D layout: VGPR v, lanes 0-15 -> M=v ; lanes 16-31 -> M=v+8 ; N=lane%16
    #pragma unroll
    for (int v = 0; v < 8; ++v)
        out[(m0 + 8 * kh + v) * GC_D + colB] = d[v];
}

extern "C" void kernel_launch(void* const* d_in, const int* in_sizes, int n_in,
                              void* d_out, int out_size, void* d_ws, size_t ws_size,
                              hipStream_t stream) {
    const float* input  = (const float*)d_in[0];
    // d_in[1] = A (unused), d_in[2] = Dinv (unused by reference)
    const float* verts  = (const float*)d_in[3];
    const int*   faces  = (const int*)d_in[4];
    const float* sigma  = (const float*)d_in[5];
    const float* fc_w   = (const float*)d_in[6];
    const float* fc_b   = (const float*)d_in[7];
    const float* nfc_w  = (const float*)d_in[8];
    const float* nfc_b  = (const float*)d_in[9];
    float* out = (float*)d_out;

    char* ws = (char*)d_ws;
    unsigned* hash  = (unsigned*)ws;                         // 1 MB
    float*    cols  = (float*)(ws + (1u << 20));             // 32 KB
    float*    S     = (float*)(ws + (1u << 20) + 32768);     // 256 B
    float*    acc   = (float*)(ws + (1u << 20) + 33024);     // 2 MB

    gc_init_ws<<<256, 256, 0, stream>>>(hash, cols, S, acc);
    gc_colsum_S<<<64, 64, 0, stream>>>(input, S);
    gc_edges<<<(6 * GC_F + 255) / 256, 256, 0, stream>>>(
        faces, verts, sigma, input, hash, cols, acc);
    gc_fused_gemm<<<GC_N / 16, 128, 0, stream>>>(
        input, fc_w, fc_b, nfc_w, nfc_b, cols, S, acc, out);
}